// ImaginationMPCV2_44341242363957
// MI455X (gfx1250) — compile-verified
//
#include <hip/hip_runtime.h>

// ---------------------------------------------------------------------------
// Double-pendulum Hamiltonian MPC rollout + trajectory cost (fp32).
// One thread per trajectory; actions staged block-wide into LDS (transposed)
// via gfx1250 GLOBAL_LOAD_ASYNC_TO_LDS_B32 (ASYNCcnt path).
// Per-step FD Hamiltonian gradients with bit-exact trig CSE and hoisted
// reciprocals (v_rcp_f32) to minimize the dependent-loop VALU count.
// ---------------------------------------------------------------------------

#define TT  100    // horizon
#define BLK 128    // trajectories per block (4 wave32)

// ---- constants mirroring the reference -------------------------------------
#define EPS_F       1e-5f
#define TWO_EPS_F   2e-5f                    // (float)(2.0 * 1e-5)
#define INV2EPS_F   (1.0f / TWO_EPS_F)       // compile-time correctly rounded
#define DT_F        0.02f
#define HALF_DT_F   0.01f                    // (float)(0.5*DT)
#define HALF_DT_FR  0.001f                   // (float)(0.5*DT*FRICTION)
#define PI_F        3.14159265358979323846f
#define TWO_PI_F    6.28318530717958647692f

__device__ __forceinline__ float angnorm(float x) {
    // (x + pi) mod 2pi - pi  with floor-mod semantics (fmod + sign fix)
    float r = fmodf(x + PI_F, TWO_PI_F);
    if (r < 0.0f) r += TWO_PI_F;
    return r - PI_F;
}

// Kinetic+potential with precomputed 1/denom and V:
//   H = (p1^2 + 2 p2^2 - 2 p1 p2 cos(dq)) * rden + V
__device__ __forceinline__ float ham_eval(float c, float rden, float V,
                                          float p1, float p2) {
    float Tk = (p1 * p1 + 2.0f * p2 * p2 - 2.0f * p1 * p2 * c) * rden;
    return Tk + V;
}

// Trig/denominator/potential bundle for one q-point.
// dq variants kept fp-exact as the reference forms them (NOT folded to dq+/-e).
struct Bundle {
    float c0, ca, cb, cc, cd;     // cos of dq, (q1+e)-q2, q1-(q2+e), (q1-e)-q2, q1-(q2-e)
    float r0, ra, rb, rc, rd;     // 1/(2*(1+sin^2)) for each dq variant
    float V00;                    // V(cos q1, cos q2)
    float V1p, V1m, V2p, V2m;     // V with q1+/-e, q2+/-e perturbed
};

__device__ __forceinline__ float rden_of(float s) {
    float denom = 2.0f * (1.0f + s * s);
    return __builtin_amdgcn_rcpf(denom);   // v_rcp_f32, ~1 ulp
}

__device__ __forceinline__ float pot(float cq1, float cq2) {
    return 10.0f * ((1.0f - cq1) + (1.0f - cq2));
}

__device__ __forceinline__ Bundle mk_bundle(float q1, float q2) {
    Bundle b;
    float dq  = q1 - q2;
    float dqa = (q1 + EPS_F) - q2;
    float dqb = q1 - (q2 + EPS_F);
    float dqc = (q1 - EPS_F) - q2;
    float dqd = q1 - (q2 - EPS_F);

    float s0 = __sinf(dq);   b.c0 = __cosf(dq);
    float sa = __sinf(dqa);  b.ca = __cosf(dqa);
    float sb = __sinf(dqb);  b.cb = __cosf(dqb);
    float sc = __sinf(dqc);  b.cc = __cosf(dqc);
    float sd = __sinf(dqd);  b.cd = __cosf(dqd);

    b.r0 = rden_of(s0);
    b.ra = rden_of(sa);
    b.rb = rden_of(sb);
    b.rc = rden_of(sc);
    b.rd = rden_of(sd);

    float cq1  = __cosf(q1);          float cq2  = __cosf(q2);
    float cq1p = __cosf(q1 + EPS_F);  float cq1m = __cosf(q1 - EPS_F);
    float cq2p = __cosf(q2 + EPS_F);  float cq2m = __cosf(q2 - EPS_F);

    b.V00 = pot(cq1, cq2);
    b.V1p = pot(cq1p, cq2);
    b.V1m = pot(cq1m, cq2);
    b.V2p = pot(cq1, cq2p);
    b.V2m = pot(cq1, cq2m);
    return b;
}

// Central-difference dH/dp (trig/denoms unchanged by p-perturbation).
__device__ __forceinline__ void p_grads(const Bundle& b, float p1, float p2,
                                        float& dp1, float& dp2) {
    dp1 = (ham_eval(b.c0, b.r0, b.V00, p1 + EPS_F, p2) -
           ham_eval(b.c0, b.r0, b.V00, p1 - EPS_F, p2)) * INV2EPS_F;
    dp2 = (ham_eval(b.c0, b.r0, b.V00, p1, p2 + EPS_F) -
           ham_eval(b.c0, b.r0, b.V00, p1, p2 - EPS_F)) * INV2EPS_F;
}

// Central-difference dH/dq.
__device__ __forceinline__ void q_grads(const Bundle& b, float p1, float p2,
                                        float& dq1, float& dq2) {
    dq1 = (ham_eval(b.ca, b.ra, b.V1p, p1, p2) -
           ham_eval(b.cc, b.rc, b.V1m, p1, p2)) * INV2EPS_F;
    dq2 = (ham_eval(b.cb, b.rb, b.V2p, p1, p2) -
           ham_eval(b.cd, b.rd, b.V2m, p1, p2)) * INV2EPS_F;
}

// One symplectic step (Stormer-Verlet with friction + torque on joint 2).
__device__ __forceinline__ void sym_step(float& q1, float& q2,
                                         float& p1, float& p2, float a) {
    Bundle b = mk_bundle(q1, q2);
    float gq1, gq2, gp1, gp2;
    q_grads(b, p1, p2, gq1, gq2);
    p_grads(b, p1, p2, gp1, gp2);

    float ph1 = p1 - HALF_DT_F * gq1 - HALF_DT_FR * gp1;
    float ph2 = p2 - HALF_DT_F * gq2 - HALF_DT_FR * gp2;

    // mid-point dH/dp: same q -> reuse bundle (zero new trans ops)
    float gm1, gm2;
    p_grads(b, ph1, ph2, gm1, gm2);

    float qn1 = q1 + DT_F * gm1;
    float qn2 = q2 + DT_F * gm2;

    Bundle be = mk_bundle(qn1, qn2);
    float eq1, eq2, ep1, ep2;
    q_grads(be, ph1, ph2, eq1, eq2);
    p_grads(be, ph1, ph2, ep1, ep2);

    p1 = ph1 - HALF_DT_F * eq1 - HALF_DT_FR * ep1;
    p2 = ph2 - HALF_DT_F * eq2 - HALF_DT_FR * ep2;
    p2 += a * DT_F;                 // torque on TORQUE_IDX = 1
    q1 = qn1;
    q2 = qn2;
}

__global__ __launch_bounds__(BLK)
void ImaginationMPCV2_kernel(const float* __restrict__ q0,
                             const float* __restrict__ p0,
                             const float* __restrict__ actions,
                             const float* __restrict__ target,
                             float* __restrict__ out, int n) {
    // Transposed action tile: sm[t*BLK + lane] -> conflict-free DS reads.
    __shared__ float sm[TT * BLK];

    const int tid = threadIdx.x;
    const int r0  = blockIdx.x * BLK;
    const int rows = (n - r0 < BLK) ? (n - r0) : BLK;
    const int total = rows * TT;
    const int gbase = r0 * TT;      // rows are contiguous in `actions`

    // ---- stage actions into LDS via gfx1250 async global->LDS DMA ----------
    for (int idx = tid; idx < total; idx += BLK) {
        int row = idx / TT;
        int col = idx - row * TT;
        unsigned long long gaddr =
            (unsigned long long)(const void*)(actions + (gbase + idx));
        // generic shared pointer: low 32 bits are the LDS byte offset
        unsigned lds_off = (unsigned)(unsigned long long)(void*)(&sm[col * BLK + row]);
        asm volatile("global_load_async_to_lds_b32 %0, %1, off"
                     :: "v"(lds_off), "v"(gaddr) : "memory");
    }
    asm volatile("s_wait_asynccnt 0" ::: "memory");
    __syncthreads();

    const int i = r0 + tid;
    if (i < n) {
        const float2 qv = ((const float2*)q0)[i];
        const float2 pv = ((const float2*)p0)[i];
        float q1 = qv.x, q2 = qv.y;
        float p1 = pv.x, p2 = pv.y;
        const float tg1 = target[0];
        const float tg2 = target[1];

        float cost = 0.0f;
        #pragma unroll 1
        for (int t = 0; t < TT; ++t) {
            float a = sm[t * BLK + tid];
            // running cost at pre-step state (q_traj[:, t], p_traj[:, t]) + control
            float e1 = angnorm(q1 - tg1);
            float e2 = angnorm(q2 - tg2);
            cost += e1 * e1 + e2 * e2 + p1 * p1 + p2 * p2 + 0.01f * (a * a);
            sym_step(q1, q2, p1, p2, a);
        }
        // terminal cost at final state
        float e1 = angnorm(q1 - tg1);
        float e2 = angnorm(q2 - tg2);
        cost += 10.0f * (e1 * e1 + e2 * e2) + (p1 * p1 + p2 * p2);

        out[i] = cost;
    }
}

extern "C" void kernel_launch(void* const* d_in, const int* in_sizes, int n_in,
                              void* d_out, int out_size, void* d_ws, size_t ws_size,
                              hipStream_t stream) {
    const float* q0      = (const float*)d_in[0];
    const float* p0      = (const float*)d_in[1];
    const float* actions = (const float*)d_in[2];
    const float* target  = (const float*)d_in[3];
    float* out = (float*)d_out;

    const int n = in_sizes[0] / 2;      // q0 is (n, 2)
    const int blocks = (n + BLK - 1) / BLK;
    hipLaunchKernelGGL(ImaginationMPCV2_kernel, dim3(blocks), dim3(BLK), 0, stream,
                       q0, p0, actions, target, out, n);
}